// SingleGAP_2980707303588
// MI455X (gfx1250) — compile-verified
//
#include <hip/hip_runtime.h>

// ---------------------------------------------------------------------------
// SingleGAP for MI455X (gfx1250), wave32.
//   feat_raw[n] = Wf(64x3) @ xknn[n](3x16) + bf      -> 4x V_WMMA_F32_16X16X4_F32
//   scores[n,k] = W1.feat_raw[:,k] + v.x[n] + const  (v = W1^T Wf precomputed)
//   softmax over axis=0 (batch): global per-column max+sum over N
//   x_attn[n,16a+b] = sum_kk xc[n,kk] * feat_raw[n, 4kk+a, b]
// Memory-bound (~1.3 GB traffic -> ~55us floor @ 23.3 TB/s); the tiny per-point
// GEMM is recomputed in each pass instead of round-tripping feat through HBM.
// ---------------------------------------------------------------------------

typedef float v2f __attribute__((ext_vector_type(2)));
typedef float v8f __attribute__((ext_vector_type(8)));

#define PTS_PER_WAVE   8
#define WAVES_PER_BLK  8
#define PTS_PER_BLK    (PTS_PER_WAVE * WAVES_PER_BLK)   // 64 points / block

// Order-preserving float<->uint encoding so atomicMax(uint) == float max.
__device__ __forceinline__ unsigned enc_f32(float f) {
  unsigned u = __float_as_uint(f);
  return (u & 0x80000000u) ? ~u : (u | 0x80000000u);
}
__device__ __forceinline__ float dec_f32(unsigned u) {
  return __uint_as_float((u & 0x80000000u) ? (u & 0x7FFFFFFFu) : ~u);
}

// Per-wave constant fragments: A = Wf tiles (16x4, d padded 3->4), C = bf bias,
// and the per-lane W1 values matching the D-fragment layout.
__device__ __forceinline__ void load_wave_frags(const float* __restrict__ Wf,
                                                const float* __restrict__ bf,
                                                const float* __restrict__ W1,
                                                int lane, v2f a[4], v8f cb[4],
                                                v8f w1v[4]) {
  const bool hi = lane >= 16;
  const int  m  = lane & 15;
#pragma unroll
  for (int t = 0; t < 4; ++t) {
    const int f = 16 * t + m;
    // A 16x4 f32 layout: VGPR0 holds K=0(lo)/K=2(hi), VGPR1 holds K=1(lo)/K=3(hi)
    a[t].x = Wf[f * 3 + (hi ? 2 : 0)];
    a[t].y = hi ? 0.0f : Wf[f * 3 + 1];
#pragma unroll
    for (int j = 0; j < 8; ++j) {
      const int fr = 16 * t + j + (hi ? 8 : 0);   // D-frag row for VGPR j
      cb[t][j]  = bf[fr];
      w1v[t][j] = W1[fr];
    }
  }
}

// One point: build B from xknn[n] (3x16, padded to 4x16), 4 WMMAs -> D tiles.
// Loads are unconditional (addresses valid for all lanes); the pad uses a
// v_cndmask select rather than an EXEC branch.
__device__ __forceinline__ void wmma_point(const float* __restrict__ xk, int lane,
                                           const v2f a[4], const v8f cb[4],
                                           v8f d[4]) {
  const bool hi = lane >= 16;
  const int  k  = lane & 15;
  const float e0 = xk[hi ? (32 + k) : k];   // d=2 : d=0
  const float e1 = xk[16 + k];              // d=1 (loaded by all lanes)
  v2f b;
  b.x = e0;
  b.y = hi ? 0.0f : e1;                     // pad row d=3 -> 0 on hi half
#pragma unroll
  for (int t = 0; t < 4; ++t)
    d[t] = __builtin_amdgcn_wmma_f32_16x16x4_f32(
        /*neg_a=*/false, a[t], /*neg_b=*/false, b,
        /*c_mod=*/(short)0, cb[t], /*reuse_a=*/false, /*reuse_b=*/false);
}

// score partial: W1 dot over this lane's 32 f-rows, fold halves with shfl_xor.
__device__ __forceinline__ float score_from_d(const v8f d[4], const v8f w1v[4]) {
  float p = 0.0f;
#pragma unroll
  for (int t = 0; t < 4; ++t)
#pragma unroll
    for (int j = 0; j < 8; ++j) p += w1v[t][j] * d[t][j];
  p += __shfl_xor(p, 16, 32);
  return p;   // = W1 . feat_raw[:, k=lane&15]  (bias bf already in D via C)
}

// Wave-uniform v.x[n] via SMEM (readfirstlane makes the index scalar).
__device__ __forceinline__ float vdotx(const float* __restrict__ x, int n,
                                       float v0, float v1, float v2) {
  const int nu = __builtin_amdgcn_readfirstlane(n);
  const float* xp = x + 3 * (size_t)nu;
  return v0 * xp[0] + v1 * xp[1] + v2 * xp[2];
}

// ---------------------------------------------------------------------------
// ws layout (floats): [0..2]=v, [3]=const0(=2*b1 + W1.bf),
//                     [4..19]=colmax (encoded uint), [20..35]=colsum
// ---------------------------------------------------------------------------

__global__ void prep_kernel(const float* __restrict__ Wf,
                            const float* __restrict__ bf,
                            const float* __restrict__ W1,
                            const float* __restrict__ b1,
                            float* __restrict__ ws) {
  if (threadIdx.x == 0) {
    float v0 = 0.f, v1 = 0.f, v2 = 0.f, wb = 0.f;
    for (int f = 0; f < 64; ++f) {
      const float w = W1[f];
      v0 += w * Wf[f * 3 + 0];
      v1 += w * Wf[f * 3 + 1];
      v2 += w * Wf[f * 3 + 2];
      wb += w * bf[f];
    }
    ws[0] = v0; ws[1] = v1; ws[2] = v2;
    ws[3] = 2.0f * b1[0] + wb;
  }
  if (threadIdx.x < 16) {
    ((unsigned*)ws)[4 + threadIdx.x] = 0u;   // enc of very-negative -> identity
    ws[20 + threadIdx.x] = 0.0f;
  }
}

// Pass 1: write feat ([F,K] row-major == reference's reinterpreted output),
//         global per-column score max.
__global__ void __launch_bounds__(256)
pass1_kernel(const float* __restrict__ xknn, const float* __restrict__ x,
             const float* __restrict__ Wf, const float* __restrict__ bf,
             const float* __restrict__ W1, float* __restrict__ feat_out,
             float* __restrict__ ws) {
  __shared__ unsigned smax[16];
  const int lane = threadIdx.x & 31;
  const int wave = threadIdx.x >> 5;
  if (threadIdx.x < 16) smax[threadIdx.x] = 0u;
  __syncthreads();

  v2f a[4]; v8f cb[4], w1v[4];
  load_wave_frags(Wf, bf, W1, lane, a, cb, w1v);
  const float v0 = ws[0], v1 = ws[1], v2 = ws[2], c0 = ws[3];

  const bool hi   = lane >= 16;
  const int  k    = lane & 15;
  const int  off  = hi ? 8 : 0;
  const int  base = blockIdx.x * PTS_PER_BLK + wave * PTS_PER_WAVE;

  float lmax = -3.402823466e38f;
#pragma unroll
  for (int i = 0; i < PTS_PER_WAVE; ++i) {
    const int n = base + i;
    const float* xk = xknn + (size_t)n * 48;
    if (i + 1 < PTS_PER_WAVE)
      __builtin_prefetch(xknn + (size_t)(n + 1) * 48, 0, 3);  // WGP-scope prefetch
    v8f d[4];
    wmma_point(xk, lane, a, cb, d);
    // feat store: addr = n*1024 + (16t + j + off)*16 + k  (coalesced 64B runs)
    float* fo = feat_out + (size_t)n * 1024 + (off * 16 + k);
#pragma unroll
    for (int t = 0; t < 4; ++t)
#pragma unroll
      for (int j = 0; j < 8; ++j) fo[t * 256 + j * 16] = d[t][j];
    const float sc = score_from_d(d, w1v) + vdotx(x, n, v0, v1, v2) + c0;
    lmax = fmaxf(lmax, sc);
  }
  atomicMax(&smax[k], enc_f32(lmax));          // ds_max_u32
  __syncthreads();
  if (threadIdx.x < 16)
    atomicMax(&((unsigned*)ws)[4 + threadIdx.x], smax[threadIdx.x]);
}

// Pass 2: colsum[k] = sum_n exp(score - colmax[k])
__global__ void __launch_bounds__(256)
pass2_kernel(const float* __restrict__ xknn, const float* __restrict__ x,
             const float* __restrict__ Wf, const float* __restrict__ bf,
             const float* __restrict__ W1, float* __restrict__ ws) {
  __shared__ float ssum[16];
  const int lane = threadIdx.x & 31;
  const int wave = threadIdx.x >> 5;
  if (threadIdx.x < 16) ssum[threadIdx.x] = 0.0f;
  __syncthreads();

  v2f a[4]; v8f cb[4], w1v[4];
  load_wave_frags(Wf, bf, W1, lane, a, cb, w1v);
  const float v0 = ws[0], v1 = ws[1], v2 = ws[2], c0 = ws[3];
  const int  k    = lane & 15;
  const float cmx = dec_f32(((const unsigned*)ws)[4 + k]);
  const int  base = blockIdx.x * PTS_PER_BLK + wave * PTS_PER_WAVE;

  float lsum = 0.0f;
#pragma unroll
  for (int i = 0; i < PTS_PER_WAVE; ++i) {
    const int n = base + i;
    if (i + 1 < PTS_PER_WAVE)
      __builtin_prefetch(xknn + (size_t)(n + 1) * 48, 0, 3);
    v8f d[4];
    wmma_point(xknn + (size_t)n * 48, lane, a, cb, d);
    const float sc = score_from_d(d, w1v) + vdotx(x, n, v0, v1, v2) + c0;
    lsum += __expf(sc - cmx);
  }
  if (lane < 16) atomicAdd(&ssum[k], lsum);    // halves duplicate -> lo half only
  __syncthreads();
  if (threadIdx.x < 16) atomicAdd(&ws[20 + threadIdx.x], ssum[threadIdx.x]);
}

// Pass 3: x_attn[n,16a+b] = sum_kk xc[n,kk] * feat_raw[n, 4kk+a, b]
__global__ void __launch_bounds__(256)
pass3_kernel(const float* __restrict__ xknn, const float* __restrict__ x,
             const float* __restrict__ Wf, const float* __restrict__ bf,
             const float* __restrict__ W1, float* __restrict__ attn_out,
             const float* __restrict__ ws) {
  const int lane = threadIdx.x & 31;
  const int wave = threadIdx.x >> 5;

  v2f a[4]; v8f cb[4], w1v[4];
  load_wave_frags(Wf, bf, W1, lane, a, cb, w1v);
  const float v0 = ws[0], v1 = ws[1], v2 = ws[2], c0 = ws[3];
  const bool hi   = lane >= 16;
  const int  k    = lane & 15;
  const float cmx = dec_f32(((const unsigned*)ws)[4 + k]);
  const float inv = 1.0f / ws[20 + k];
  const int  base = blockIdx.x * PTS_PER_BLK + wave * PTS_PER_WAVE;

#pragma unroll
  for (int i = 0; i < PTS_PER_WAVE; ++i) {
    const int n = base + i;
    if (i + 1 < PTS_PER_WAVE)
      __builtin_prefetch(xknn + (size_t)(n + 1) * 48, 0, 3);
    v8f d[4];
    wmma_point(xknn + (size_t)n * 48, lane, a, cb, d);
    const float sc  = score_from_d(d, w1v) + vdotx(x, n, v0, v1, v2) + c0;
    const float xcl = __expf(sc - cmx) * inv;  // lanes 0..15 hold xc[k=lane]

    float acc0 = 0.f, acc1 = 0.f, acc2 = 0.f, acc3 = 0.f;
#pragma unroll
    for (int t = 0; t < 4; ++t) {
      // lo half: j=a -> kk=4t,   j=a+4 -> kk=4t+1
      // hi half: j=a -> kk=4t+2, j=a+4 -> kk=4t+3
      const float xc0 = __shfl(xcl, 4 * t + (hi ? 2 : 0), 32);
      const float xc1 = __shfl(xcl, 4 * t + (hi ? 3 : 1), 32);
      acc0 += xc0 * d[t][0] + xc1 * d[t][4];
      acc1 += xc0 * d[t][1] + xc1 * d[t][5];
      acc2 += xc0 * d[t][2] + xc1 * d[t][6];
      acc3 += xc0 * d[t][3] + xc1 * d[t][7];
    }
    acc0 += __shfl_xor(acc0, 16, 32);
    acc1 += __shfl_xor(acc1, 16, 32);
    acc2 += __shfl_xor(acc2, 16, 32);
    acc3 += __shfl_xor(acc3, 16, 32);
    float* ao = attn_out + (size_t)n * 64 + k;
    if (!hi) { ao[0]  = acc0; ao[16] = acc1; }
    else     { ao[32] = acc2; ao[48] = acc3; }
  }
}

extern "C" void kernel_launch(void* const* d_in, const int* in_sizes, int n_in,
                              void* d_out, int out_size, void* d_ws, size_t ws_size,
                              hipStream_t stream) {
  const float* xknn = (const float*)d_in[0];
  const float* x    = (const float*)d_in[1];
  const float* Wf   = (const float*)d_in[2];
  const float* bf   = (const float*)d_in[3];
  const float* W1   = (const float*)d_in[4];
  const float* b1   = (const float*)d_in[5];
  // d_in[6] = KNN_size (==16, baked into the kernels)

  const int n = in_sizes[0] / 48;                    // N = 262144
  float* attn_out = (float*)d_out;                   // [N, 64]
  float* feat_out = (float*)d_out + (size_t)n * 64;  // [N,16,64] (reinterp of [N,64,16])
  float* ws = (float*)d_ws;                          // 36 floats used

  const int blocks = n / PTS_PER_BLK;                // 4096

  prep_kernel<<<1, 32, 0, stream>>>(Wf, bf, W1, b1, ws);
  pass1_kernel<<<blocks, 256, 0, stream>>>(xknn, x, Wf, bf, W1, feat_out, ws);
  pass2_kernel<<<blocks, 256, 0, stream>>>(xknn, x, Wf, bf, W1, ws);
  pass3_kernel<<<blocks, 256, 0, stream>>>(xknn, x, Wf, bf, W1, attn_out, ws);
}